// MultiHeadAttention_76244259438975
// MI455X (gfx1250) — compile-verified
//
#include <hip/hip_runtime.h>

// MHA forward: B=2, L=2048, D_MODEL=1024, H=16, Dk=Dv=64.
// Fused flash-style kernel: per (b,h,16-row q-block), scores are computed with
// f32 WMMA into LDS (128KB strip fits CDNA5's 320KB WGP LDS), softmax'd in
// LDS, written once to the attn output, and consumed directly for attn@V.
// Cuts HBM traffic from ~2.2GB to ~0.62GB (attn write is compulsory).

typedef __attribute__((ext_vector_type(2))) float v2f;
typedef __attribute__((ext_vector_type(8))) float v8f;

#define B_  2
#define L_  2048
#define DM  1024
#define H_  16
#define DK  64
#define SROW 2052   // LDS row stride in floats (2048 + 4 pad: de-conflicts banks)

static __device__ __forceinline__ v8f wmma4(v2f a, v2f b, v8f acc) {
  return __builtin_amdgcn_wmma_f32_16x16x4_f32(
      false, a, false, b, (short)0, acc, false, false);
}

// One wave computes a 16x16 f32 C tile: C += A(16xK row-major, lda) * B(Kx16),
// B element (k,n) at Bp[k*bK + n*bN] (handles normal and transposed B).
static __device__ __forceinline__ v8f tile_mma_f32(
    const float* __restrict__ A, int lda,
    const float* __restrict__ Bp, int bK, int bN,
    int K, int lane, v8f acc)
{
  const int mr = lane & 15;
  const int kk = (lane >> 4) << 1;
  const float* ap = A  + (size_t)mr * lda + kk;
  const float* bp = Bp + (size_t)mr * bN  + (size_t)kk * bK;
  for (int k = 0; k < K; k += 4) {
    v2f a, b;
    a.x = ap[0];  a.y = ap[1];
    b.x = bp[0];  b.y = bp[bK];
    acc = wmma4(a, b, acc);
    ap += 4;
    bp += (size_t)4 * bK;
  }
  return acc;
}

// ---- Kernel 1: fused QKV projections, written head-split [B,H,L,DK] ----
__global__ void qkv_proj_kernel(
    const float* __restrict__ xq, const float* __restrict__ xk, const float* __restrict__ xv,
    const float* __restrict__ Wq, const float* __restrict__ Wk, const float* __restrict__ Wv,
    const float* __restrict__ bq, const float* __restrict__ bk, const float* __restrict__ bv,
    float* __restrict__ qh, float* __restrict__ kh, float* __restrict__ vh)
{
  const int lane = threadIdx.x & 31;
  const int gid  = blockIdx.x * (blockDim.x >> 5) + (threadIdx.x >> 5);
  const int ntn  = DM / 16;                 // 64
  const int per  = (B_ * L_ / 16) * ntn;    // 16384 tiles per matrix
  const int which = gid / per;
  const int rem   = gid - which * per;
  const int mt    = rem / ntn;
  const int nt    = rem - mt * ntn;

  const float* X    = which == 0 ? xq : (which == 1 ? xk : xv);
  const float* W    = which == 0 ? Wq : (which == 1 ? Wk : Wv);
  const float* bias = which == 0 ? bq : (which == 1 ? bk : bv);
  float* dstbase    = which == 0 ? qh : (which == 1 ? kh : vh);

  v8f acc = {0.f,0.f,0.f,0.f,0.f,0.f,0.f,0.f};
  acc = tile_mma_f32(X + (size_t)(mt * 16) * DM, DM, W + nt * 16, DM, 1, DM, lane, acc);

  const int colg = nt * 16 + (lane & 15);
  const int h = colg >> 6, d = colg & 63;
  const int m0 = mt * 16;
  const int bb = m0 >> 11;
  const int l0 = m0 & (L_ - 1);
  const float badd  = bias[colg];
  const int   rbase = (lane >> 4) << 3;
  float* dst = dstbase + (((size_t)(bb * H_ + h)) * L_ + l0) * DK + d;
#pragma unroll
  for (int v = 0; v < 8; ++v)
    dst[(size_t)(rbase + v) * DK] = acc[v] + badd;
}

// ---- Kernel 2: fused scores -> softmax -> attn write -> attn@V ----
// Grid: one block (256 thr = 8 waves) per (b*H+h, qt). LDS holds the [16x2048]
// score strip.
__global__ void __launch_bounds__(256)
fused_attn_kernel(
    const float* __restrict__ qh, const float* __restrict__ kh,
    const float* __restrict__ vh, float* __restrict__ attn,
    float* __restrict__ oh)
{
  __shared__ float sc[16 * SROW];           // 131,328 B score strip
  __shared__ float rowinv[16];
  __shared__ float part[4][16][16];         // partial C tiles for K-split reduce

  const int bh = blockIdx.x >> 7;           // 0..31 = b*16+h
  const int qt = blockIdx.x & 127;
  const int t    = threadIdx.x;
  const int w    = t >> 5;                  // wave 0..7
  const int lane = t & 31;
  const int mr    = lane & 15;
  const int kk    = (lane >> 4) << 1;       // 0 or 2
  const int rbase = (lane >> 4) << 3;       // 0 or 8

  // ---- Phase 1: scores strip into LDS. A fragments preloaded once. ----
  const float* Aq = qh + ((size_t)bh * L_ + qt * 16) * DK;
  v2f afrag[16];
#pragma unroll
  for (int s = 0; s < 16; ++s) {
    afrag[s].x = Aq[mr * DK + 4 * s + kk];
    afrag[s].y = Aq[mr * DK + 4 * s + kk + 1];
  }
  for (int j = 0; j < 16; ++j) {
    const int kt = w + 8 * j;               // 0..127, wave-uniform
    // B = K^T: element (k,n) = kh[(kt*16+n)*DK + k]
    const float* bp = kh + ((size_t)bh * L_ + kt * 16) * DK + mr * DK + kk;
    v8f acc = {0.f,0.f,0.f,0.f,0.f,0.f,0.f,0.f};
#pragma unroll
    for (int s = 0; s < 16; ++s) {
      v2f b; b.x = bp[0]; b.y = bp[1];
      acc = wmma4(afrag[s], b, acc);
      bp += 4;
    }
#pragma unroll
    for (int v = 0; v < 8; ++v)
      sc[(rbase + v) * SROW + kt * 16 + mr] = acc[v] * 0.125f;
  }
  __syncthreads();

  // ---- Phase 2: row softmax in LDS (16 threads per row) ----
  {
    const int r  = t >> 4;                  // 0..15
    const int c0 = t & 15;
    float m = -3.4e38f;
    for (int j = 0; j < 128; ++j)
      m = fmaxf(m, sc[r * SROW + c0 + 16 * j]);
#pragma unroll
    for (int off = 8; off > 0; off >>= 1)
      m = fmaxf(m, __shfl_xor(m, off, 16));
    float s = 0.f;
    for (int j = 0; j < 128; ++j) {
      const int idx = r * SROW + c0 + 16 * j;
      const float e = __expf(sc[idx] - m);
      sc[idx] = e;
      s += e;
    }
#pragma unroll
    for (int off = 8; off > 0; off >>= 1)
      s += __shfl_xor(s, off, 16);
    if (c0 == 0) rowinv[r] = 1.0f / s;
  }
  __syncthreads();

  // ---- Phase 3: normalize in LDS + single coalesced write of attn ----
  {
    float* arow = attn + ((size_t)bh * L_ + qt * 16) * (size_t)L_;
    for (int i = t; i < 16 * L_; i += 256) {
      const int rr = i >> 11, cc = i & (L_ - 1);
      const float val = sc[rr * SROW + cc] * rowinv[rr];
      sc[rr * SROW + cc] = val;
      arow[(size_t)rr * L_ + cc] = val;
    }
  }
  __syncthreads();

  // ---- Phase 4: O tile [16x64] = sc[16x2048] @ V[2048x64] ----
  // wave w: output d-tile (w&3), K-half (w>>2); partials reduced via LDS.
  {
    const int dt = w & 3, khalf = w >> 2;
    const int kbase = khalf * 1024;
    const float* bp = vh + (size_t)bh * L_ * DK + dt * 16
                    + (size_t)(kbase + kk) * DK + mr;   // (k,n) = vh[k*DK+n]
    v8f acc = {0.f,0.f,0.f,0.f,0.f,0.f,0.f,0.f};
    for (int k = 0; k < 1024; k += 4) {
      v2f a, b;
      const int kg = kbase + k + kk;
      a.x = sc[mr * SROW + kg];
      a.y = sc[mr * SROW + kg + 1];
      b.x = bp[0];
      b.y = bp[DK];
      acc = wmma4(a, b, acc);
      bp += 4 * DK;
    }
    if (w >= 4) {
#pragma unroll
      for (int v = 0; v < 8; ++v)
        part[dt][rbase + v][mr] = acc[v];
    }
    __syncthreads();
    if (w < 4) {
      const int bb = bh >> 4, h = bh & 15;
      float* dst = oh + ((size_t)(bb * L_ + qt * 16)) * DM + h * DK + dt * 16 + mr;
#pragma unroll
      for (int v = 0; v < 8; ++v)
        dst[(size_t)(rbase + v) * DM] = acc[v] + part[dt][rbase + v][mr];
    }
  }
}

// ---- Kernel 3: out = out_heads @ Wo + bo ----
__global__ void oproj_kernel(
    const float* __restrict__ oh, const float* __restrict__ Wo,
    const float* __restrict__ bo, float* __restrict__ out)
{
  const int lane = threadIdx.x & 31;
  const int gid  = blockIdx.x * (blockDim.x >> 5) + (threadIdx.x >> 5);
  const int nt = gid & 63;
  const int mt = gid >> 6;

  v8f acc = {0.f,0.f,0.f,0.f,0.f,0.f,0.f,0.f};
  acc = tile_mma_f32(oh + (size_t)(mt * 16) * DM, DM, Wo + nt * 16, DM, 1, DM, lane, acc);

  const int col = nt * 16 + (lane & 15);
  const float badd = bo[col];
  const int rbase = (lane >> 4) << 3;
  float* dst = out + (size_t)(mt * 16) * DM + col;
#pragma unroll
  for (int v = 0; v < 8; ++v)
    dst[(size_t)(rbase + v) * DM] = acc[v] + badd;
}

extern "C" void kernel_launch(void* const* d_in, const int* in_sizes, int n_in,
                              void* d_out, int out_size, void* d_ws, size_t ws_size,
                              hipStream_t stream) {
  const float* q  = (const float*)d_in[0];
  const float* k  = (const float*)d_in[1];
  const float* v  = (const float*)d_in[2];
  const float* Wq = (const float*)d_in[3];
  const float* bq = (const float*)d_in[4];
  const float* Wk = (const float*)d_in[5];
  const float* bk = (const float*)d_in[6];
  const float* Wv = (const float*)d_in[7];
  const float* bv = (const float*)d_in[8];
  const float* Wo = (const float*)d_in[9];
  const float* bo = (const float*)d_in[10];

  float* out  = (float*)d_out;                       // [B, L, DM]
  float* attn = out + (size_t)B_ * L_ * DM;          // [B, H, L, L]

  const size_t Q = (size_t)B_ * H_ * L_ * DK;        // 4,194,304 elems each
  float* w  = (float*)d_ws;                          // needs 4*Q floats (~67MB)
  float* qh = w;
  float* kh = w + Q;
  float* vh = w + 2 * Q;
  float* oh = w + 3 * Q;

  // 1) QKV projections: 3 * 256 * 64 = 49152 tiles, 4 waves/block
  qkv_proj_kernel<<<12288, 128, 0, stream>>>(q, k, v, Wq, Wk, Wv, bq, bk, bv,
                                             qh, kh, vh);
  // 2) fused attention: one block per (b*H+h, qt) = 32*128 = 4096 blocks
  fused_attn_kernel<<<4096, 256, 0, stream>>>(qh, kh, vh, attn, oh);
  // 3) output projection: 256*64 = 16384 tiles, 4 waves/block
  oproj_kernel<<<4096, 128, 0, stream>>>(oh, Wo, bo, out);
}